// BEV_Offset_pred_head_cfg_29472065585885
// MI455X (gfx1250) — compile-verified
//
#include <hip/hip_runtime.h>

typedef float v2f __attribute__((ext_vector_type(2)));
typedef float v8f __attribute__((ext_vector_type(8)));

#define Bn    2
#define FEA   128
#define Hn    480
#define Wn    360
#define NHn   32
#define PTn   32
#define EMBn  3
#define Nn    120000
#define NPTS  (Bn * Nn)          // 240000
#define NBINS (Bn * NHn)         // 64
#define MAXTILES (NPTS / 16 + NBINS)
#define BN_EPS 1e-5f

// ---------------- binning kernels ----------------

__global__ void k_init(int* __restrict__ bin_count) {
    int t = threadIdx.x;
    if (t < NBINS) bin_count[t] = 0;
}

__global__ void k_count(const int* __restrict__ gi, int* __restrict__ bin_count) {
    int p = blockIdx.x * blockDim.x + threadIdx.x;
    if (p >= NPTS) return;
    int b  = p / Nn;
    int g2 = gi[p * 3 + 2] & (NHn - 1);
    atomicAdd(&bin_count[b * NHn + g2], 1);
}

__global__ void k_scan(const int* __restrict__ bin_count,
                       int* __restrict__ poff, int* __restrict__ cursor) {
    int off = 0;
    for (int i = 0; i < NBINS; ++i) {
        poff[i]   = off;
        cursor[i] = off;
        off += (bin_count[i] + 15) & ~15;   // pad each bin to a multiple of 16
    }
    poff[NBINS] = off;
}

__global__ void k_scatter(const int* __restrict__ gi,
                          int* __restrict__ cursor, int* __restrict__ order) {
    int p = blockIdx.x * blockDim.x + threadIdx.x;
    if (p >= NPTS) return;
    int b  = p / Nn;
    int g2 = gi[p * 3 + 2] & (NHn - 1);
    int pos = atomicAdd(&cursor[b * NHn + g2], 1);
    order[pos] = p;
}

// ---------------- fused gather-conv + MLP kernel ----------------
// One wave32 per 16-point tile. All points in a tile share (b, nh), so the
// 32x128 conv_w slice is a shared B matrix -> clean f32 WMMA GEMM.

__global__ __launch_bounds__(32) void k_gemm(
    const float* __restrict__ x,       const int*   __restrict__ gi,
    const float* __restrict__ conv_w,  const float* __restrict__ conv_b,
    const float* __restrict__ w1,      const float* __restrict__ b1,
    const float* __restrict__ gmm,     const float* __restrict__ bta,
    const float* __restrict__ mean,    const float* __restrict__ var,
    const float* __restrict__ w2,      const float* __restrict__ b2,
    const int*   __restrict__ poff,    const int*   __restrict__ bin_count,
    const int*   __restrict__ order,
    float* __restrict__ out_off,       float* __restrict__ out_fea)
{
    __shared__ float wS[PTn * (FEA + 1)];   // weight slice, padded stride
    __shared__ float xS[16 * (FEA + 1)];    // gathered x vectors
    __shared__ float feaS[16 * 33];         // pt_fea tile (D-layout -> A-layout relay)
    __shared__ float hS[16 * 33];           // post BN/ReLU hidden tile
    __shared__ int   pidS[16];
    __shared__ int   boffS[16];

    const int lane = threadIdx.x;
    const int l16  = lane & 15;
    const int half = lane >> 4;             // 0: K pair {0,1}, 1: K pair {2,3}

    const int base  = blockIdx.x * 16;
    const int total = poff[NBINS];
    if (base >= total) return;

    int bin = 0;
    for (int i = 1; i < NBINS; ++i)
        if (poff[i] <= base) bin = i;       // poff is monotonic

    const int b     = bin / NHn;
    const int nh    = bin % NHn;
    const int cnt   = bin_count[bin];
    const int start = poff[bin];
    int nvalid = start + cnt - base;
    nvalid = nvalid < 16 ? nvalid : 16;

    if (lane < 16) {
        int pid = -1, boff = 0;
        if (lane < nvalid) {
            pid = order[base + lane];
            int g0 = gi[pid * 3 + 0];
            int g1 = gi[pid * 3 + 1];
            boff = (b * FEA) * (Hn * Wn) + g0 * Wn + g1;
        }
        pidS[lane]  = pid;
        boffS[lane] = boff;
    }

    // Stage the 32x128 conv_w slice (contiguous 16 KB -> coalesced)
    const float* cw = conv_w + (nh * PTn) * FEA;
    for (int i = lane; i < PTn * FEA; i += 32) {
        int o = i >> 7, c = i & 127;
        wS[o * (FEA + 1) + c] = cw[i];
    }
    __syncthreads();

    // Stage gathered x vectors (scattered 4B loads, L2-hot)
    for (int m = 0; m < 16; ++m) {
        int pid  = pidS[m];
        int boff = boffS[m];
        for (int c = lane; c < FEA; c += 32)
            xS[m * (FEA + 1) + c] = (pid >= 0) ? x[boff + c * (Hn * Wn)] : 0.f;
    }
    __syncthreads();

    // ---- conv GEMM: pt_fea(16x32) = X(16x128) x Wslice^T, K=128, f32 WMMA ----
    v8f acc0 = {}; v8f acc1 = {};
    for (int k = 0; k < FEA; k += 4) {
        const int kk = k + 2 * half;
        v2f a;   a.x   = xS[l16 * (FEA + 1) + kk];        a.y   = xS[l16 * (FEA + 1) + kk + 1];
        v2f bb0; bb0.x = wS[l16 * (FEA + 1) + kk];        bb0.y = wS[l16 * (FEA + 1) + kk + 1];
        v2f bb1; bb1.x = wS[(l16 + 16) * (FEA + 1) + kk]; bb1.y = wS[(l16 + 16) * (FEA + 1) + kk + 1];
        acc0 = __builtin_amdgcn_wmma_f32_16x16x4_f32(false, a, false, bb0, (short)0, acc0, false, false);
        acc1 = __builtin_amdgcn_wmma_f32_16x16x4_f32(false, a, false, bb1, (short)0, acc1, false, false);
    }

    // + conv bias (column == lane -> per-lane scalar), emit pt_fea, relay via LDS
    const float bias0 = conv_b[nh * PTn + l16];
    const float bias1 = conv_b[nh * PTn + l16 + 16];
    #pragma unroll
    for (int r = 0; r < 8; ++r) {
        int m = r + 8 * half;
        float v0 = acc0[r] + bias0;
        float v1 = acc1[r] + bias1;
        feaS[m * 33 + l16]      = v0;
        feaS[m * 33 + l16 + 16] = v1;
        int pid = pidS[m];
        if (pid >= 0) {
            out_fea[pid * PTn + l16]      = v0;
            out_fea[pid * PTn + l16 + 16] = v1;
        }
    }
    __syncthreads();

    // ---- MLP layer 1: h(16x32) = pt_fea(16x32) x w1^T(32x32), K=32 ----
    v8f h0 = {}; v8f h1 = {};
    for (int k = 0; k < PTn; k += 4) {
        const int kk = k + 2 * half;
        v2f a;   a.x   = feaS[l16 * 33 + kk];        a.y   = feaS[l16 * 33 + kk + 1];
        v2f bb0; bb0.x = w1[l16 * PTn + kk];         bb0.y = w1[l16 * PTn + kk + 1];
        v2f bb1; bb1.x = w1[(l16 + 16) * PTn + kk];  bb1.y = w1[(l16 + 16) * PTn + kk + 1];
        h0 = __builtin_amdgcn_wmma_f32_16x16x4_f32(false, a, false, bb0, (short)0, h0, false, false);
        h1 = __builtin_amdgcn_wmma_f32_16x16x4_f32(false, a, false, bb1, (short)0, h1, false, false);
    }

    // BN (running stats) + ReLU; channel == column -> per-lane constants.
    // h = (acc + b1[c] - mean[c]) * rsqrt(var+eps)*gamma[c] + beta[c]
    {
        const int c0 = l16, c1 = l16 + 16;
        float s0 = rsqrtf(var[c0] + BN_EPS) * gmm[c0];
        float s1 = rsqrtf(var[c1] + BN_EPS) * gmm[c1];
        float t0 = bta[c0] + (b1[c0] - mean[c0]) * s0;
        float t1 = bta[c1] + (b1[c1] - mean[c1]) * s1;
        #pragma unroll
        for (int r = 0; r < 8; ++r) {
            int m = r + 8 * half;
            float u0 = h0[r] * s0 + t0; u0 = u0 > 0.f ? u0 : 0.f;
            float u1 = h1[r] * s1 + t1; u1 = u1 > 0.f ? u1 : 0.f;
            hS[m * 33 + l16]      = u0;
            hS[m * 33 + l16 + 16] = u1;
        }
    }
    __syncthreads();

    // ---- final projection to EMB=3 (tiny; scalar per point) ----
    if (lane < 16) {
        int pid = pidS[lane];
        if (pid >= 0) {
            #pragma unroll
            for (int e = 0; e < EMBn; ++e) {
                float s = b2[e];
                #pragma unroll
                for (int c = 0; c < PTn; ++c)
                    s += hS[lane * 33 + c] * w2[e * PTn + c];
                out_off[pid * EMBn + e] = s;
            }
        }
    }
}

// ---------------- host launcher ----------------

extern "C" void kernel_launch(void* const* d_in, const int* in_sizes, int n_in,
                              void* d_out, int out_size, void* d_ws, size_t ws_size,
                              hipStream_t stream) {
    const float* x      = (const float*)d_in[0];
    const int*   gi     = (const int*)  d_in[1];
    const float* conv_w = (const float*)d_in[2];
    const float* conv_b = (const float*)d_in[3];
    const float* w1     = (const float*)d_in[4];
    const float* b1     = (const float*)d_in[5];
    const float* gmm    = (const float*)d_in[6];
    const float* bta    = (const float*)d_in[7];
    const float* mean   = (const float*)d_in[8];
    const float* var    = (const float*)d_in[9];
    const float* w2     = (const float*)d_in[10];
    const float* b2     = (const float*)d_in[11];

    float* out     = (float*)d_out;
    float* out_off = out;                  // offsets [B,N,3]
    float* out_fea = out + NPTS * EMBn;    // pt_fea  [B,N,32]

    int* wsI       = (int*)d_ws;
    int* bin_count = wsI;                          // 64
    int* poff      = wsI + NBINS;                  // 65
    int* cursor    = wsI + NBINS + NBINS + 1;      // 64
    int* order     = cursor + NBINS;               // NPTS

    k_init   <<<1, NBINS, 0, stream>>>(bin_count);
    k_count  <<<(NPTS + 255) / 256, 256, 0, stream>>>(gi, bin_count);
    k_scan   <<<1, 1, 0, stream>>>(bin_count, poff, cursor);
    k_scatter<<<(NPTS + 255) / 256, 256, 0, stream>>>(gi, cursor, order);
    k_gemm   <<<MAXTILES, 32, 0, stream>>>(x, gi, conv_w, conv_b, w1, b1,
                                           gmm, bta, mean, var, w2, b2,
                                           poff, bin_count, order,
                                           out_off, out_fea);
}